// Box_diamond_57071525429385
// MI455X (gfx1250) — compile-verified
//
#include <hip/hip_runtime.h>
#include <cstdint>

#define PN        512      // number of p-units
#define ROW_F     4096     // floats per row of x
#define NROWS     4096     // B
#define RPB       16       // rows per block (amortizes W1/W2 register load)
#define THREADS   512

// ---------------------------------------------------------------------------
// Precompute W1[p][m][l] and W2[p][l] from t0,t1,t2 (depends only on params).
// ---------------------------------------------------------------------------
__global__ void bd_precompute_w(const float* __restrict__ t0,
                                const float* __restrict__ t1,
                                const float* __restrict__ t2,
                                float* __restrict__ W1,    // [512][8][8]
                                float* __restrict__ W2) {  // [512][8]
    int p = blockIdx.x * blockDim.x + threadIdx.x;
    if (p >= PN) return;
    float c0 = t0[p], c1 = t1[p], c2 = t2[p];
#pragma unroll
    for (int m = 0; m < 8; ++m) {
#pragma unroll
        for (int l = 0; l < 8; ++l) {
            float z = (float)(l - m) * ((float)m + c2 - (float)l);
            W1[p * 64 + m * 8 + l] = __builtin_amdgcn_rcpf(1.0f + __expf(-z));
        }
    }
#pragma unroll
    for (int l = 0; l < 8; ++l) {
        float za = ((float)l - c0) * (c1 - (float)l);
        float zb = (7.0f - c2 - (float)l) * (float)l;
        float sa = __builtin_amdgcn_rcpf(1.0f + __expf(-za));
        float sb = __builtin_amdgcn_rcpf(1.0f + __expf(-zb));
        W2[p * 8 + l] = sa * sb;
    }
}

// ---------------------------------------------------------------------------
// Async stage one 16KB row of x into LDS: 512 threads x 2 b128 each.
// ---------------------------------------------------------------------------
__device__ __forceinline__ void bd_async_row(const float* grow, unsigned lds_base, int tid) {
    const float* g0 = grow + tid * 4;
    const float* g1 = grow + (tid + 512) * 4;
    unsigned l0 = lds_base + tid * 16u;
    unsigned l1 = lds_base + (tid + 512) * 16u;
    asm volatile("global_load_async_to_lds_b128 %0, %1, off"
                 :: "v"(l0), "v"(g0) : "memory");
    asm volatile("global_load_async_to_lds_b128 %0, %1, off"
                 :: "v"(l1), "v"(g1) : "memory");
}

// ---------------------------------------------------------------------------
// Main kernel: thread p computes out[b][p] for RPB rows, double-buffered
// async LDS staging of x rows (ASYNCcnt pipeline).
// ---------------------------------------------------------------------------
__global__ void __launch_bounds__(THREADS)
bd_main(const float* __restrict__ x,
        const float* __restrict__ W1,
        const float* __restrict__ W2,
        float* __restrict__ out) {
    __shared__ float ldsx[2][ROW_F];

    const int tid  = threadIdx.x;                 // == p
    const int row0 = blockIdx.x * RPB;

    // per-thread weights live in VGPRs for all RPB rows
    float w1[64], w2[8];
#pragma unroll
    for (int i = 0; i < 64; ++i) w1[i] = W1[tid * 64 + i];
#pragma unroll
    for (int i = 0; i < 8; ++i)  w2[i] = W2[tid * 8 + i];

    const unsigned lds0 = (unsigned)(unsigned long long)(&ldsx[0][0]);
    const unsigned lds1 = (unsigned)(unsigned long long)(&ldsx[1][0]);

    // prime the pipeline with row 0
    bd_async_row(x + (size_t)row0 * ROW_F, lds0, tid);

    const int xbase = (tid >> 2) * 32 + (tid & 3);  // word offset inside a row

#pragma unroll 2
    for (int r = 0; r < RPB; ++r) {
        if (r + 1 < RPB) {
            // kick off next row into the other buffer, then wait for current
            bd_async_row(x + (size_t)(row0 + r + 1) * ROW_F,
                         ((r + 1) & 1) ? lds1 : lds0, tid);
            asm volatile("s_wait_asynccnt 0x2" ::: "memory");
        } else {
            asm volatile("s_wait_asynccnt 0x0" ::: "memory");
        }
        __syncthreads();   // all waves' loads for row r are complete

        const float* xrow = (r & 1) ? &ldsx[1][0] : &ldsx[0][0];
        float xr[8];
#pragma unroll
        for (int l = 0; l < 8; ++l) xr[l] = xrow[xbase + l * 4];

        // stage 1: y1[m] = 1/(1 - log(prod_l (1 - xr[l]*W1[m][l])))
        float y1[8];
#pragma unroll
        for (int m = 0; m < 8; ++m) {
            float a0 = __builtin_fmaf(-xr[0], w1[m * 8 + 0], 1.0f);
            float a1 = __builtin_fmaf(-xr[1], w1[m * 8 + 1], 1.0f);
            float a2 = __builtin_fmaf(-xr[2], w1[m * 8 + 2], 1.0f);
            float a3 = __builtin_fmaf(-xr[3], w1[m * 8 + 3], 1.0f);
            float a4 = __builtin_fmaf(-xr[4], w1[m * 8 + 4], 1.0f);
            float a5 = __builtin_fmaf(-xr[5], w1[m * 8 + 5], 1.0f);
            float a6 = __builtin_fmaf(-xr[6], w1[m * 8 + 6], 1.0f);
            float a7 = __builtin_fmaf(-xr[7], w1[m * 8 + 7], 1.0f);
            float prod = ((a0 * a1) * (a2 * a3)) * ((a4 * a5) * (a6 * a7));
            y1[m] = __builtin_amdgcn_rcpf(1.0f - __logf(prod));
        }

        // stage 2: out = 1/(1 - log(prod_l (1 - y1[l]*W2[l])))
        float b0 = __builtin_fmaf(-y1[0], w2[0], 1.0f);
        float b1 = __builtin_fmaf(-y1[1], w2[1], 1.0f);
        float b2 = __builtin_fmaf(-y1[2], w2[2], 1.0f);
        float b3 = __builtin_fmaf(-y1[3], w2[3], 1.0f);
        float b4 = __builtin_fmaf(-y1[4], w2[4], 1.0f);
        float b5 = __builtin_fmaf(-y1[5], w2[5], 1.0f);
        float b6 = __builtin_fmaf(-y1[6], w2[6], 1.0f);
        float b7 = __builtin_fmaf(-y1[7], w2[7], 1.0f);
        float prod2 = ((b0 * b1) * (b2 * b3)) * ((b4 * b5) * (b6 * b7));
        float result = __builtin_amdgcn_rcpf(1.0f - __logf(prod2));
        __builtin_nontemporal_store(result, &out[(size_t)(row0 + r) * PN + tid]);

        __syncthreads();   // everyone done reading this buffer before reuse
    }
}

// ---------------------------------------------------------------------------
extern "C" void kernel_launch(void* const* d_in, const int* in_sizes, int n_in,
                              void* d_out, int out_size, void* d_ws, size_t ws_size,
                              hipStream_t stream) {
    const float* x  = (const float*)d_in[0];
    const float* t0 = (const float*)d_in[1];
    const float* t1 = (const float*)d_in[2];
    const float* t2 = (const float*)d_in[3];

    float* W1 = (float*)d_ws;            // 512*64 floats = 128KB
    float* W2 = W1 + PN * 64;            // 512*8 floats  = 16KB

    bd_precompute_w<<<2, 256, 0, stream>>>(t0, t1, t2, W1, W2);
    bd_main<<<NROWS / RPB, THREADS, 0, stream>>>(x, W1, W2, (float*)d_out);
}